// BBoxWindowAttention_58858231824400
// MI455X (gfx1250) — compile-verified
//
#include <hip/hip_runtime.h>

// ---------------------------------------------------------------------------
// BBoxWindowAttention for MI455X (gfx1250), fp32 end-to-end using the native
// CDNA5 fp32 matrix op V_WMMA_F32_16X16X4_F32 (wave32).
//
// Shapes: B=8, T=4097, D=512, H=8, head_dim=64, WIN=64, nw=64.
// ---------------------------------------------------------------------------

typedef __attribute__((ext_vector_type(2))) float v2f;
typedef __attribute__((ext_vector_type(8))) float v8f;

#define B_   8
#define T_   4097
#define Dm   512
#define H_   8
#define HD   64
#define NW   64
#define QKVN 1536
#define SCALE 0.125f   // 1/sqrt(64)

static __device__ __forceinline__ v8f wmma_f32(v2f a, v2f b, v8f c) {
  // 8-arg form: (neg_a, A, neg_b, B, c_mod, C, reuse_a, reuse_b)
  return __builtin_amdgcn_wmma_f32_16x16x4_f32(false, a, false, b,
                                               (short)0, c, false, false);
}

// ---------------------------------------------------------------------------
// C[M,N] = A[M,K] * Bw[K,N], all row-major fp32. K % 32 == 0, N % 64 == 0.
// Block: 256 threads = 8 waves arranged 4x2; block tile 128x64.
// Each wave owns a 32x32 quad (2x2 tiles of 16x16): 4 accumulators,
// 2 A-fragments + 2 B-fragments per K-step feed 4 WMMAs.
// B is staged TRANSPOSED in LDS so each fragment is a contiguous dword pair
// (clean ds_load_2addr offset1:1, no register repacking).
// ---------------------------------------------------------------------------
__global__ __launch_bounds__(256) void gemm_f32_wmma(
    const float* __restrict__ A, const float* __restrict__ Bw,
    float* __restrict__ C, int M, int N, int K)
{
  __shared__ float As [128][33];   // As[m][k]
  __shared__ float BsT[64][33];    // BsT[n][k] = B[k][n]

  const int tid  = threadIdx.x;
  const int lane = tid & 31;
  const int wave = tid >> 5;
  const int half = lane >> 4;   // 0..1 (lane group)
  const int l16  = lane & 15;

  const int wr = wave >> 1;     // 0..3 : 32-row group
  const int wc = wave & 1;      // 0..1 : 32-col group

  const int m0b = blockIdx.x * 128;
  const int n0  = blockIdx.y * 64;

  v8f acc[4];                    // [mt*2+nt]
#pragma unroll
  for (int t = 0; t < 4; ++t)
    acc[t] = (v8f){0.f, 0.f, 0.f, 0.f, 0.f, 0.f, 0.f, 0.f};

  for (int k0 = 0; k0 < K; k0 += 32) {
    // ---- stage A tile: 128 rows x 32 cols = 1024 float4, 4 per thread ----
#pragma unroll
    for (int j = 0; j < 4; ++j) {
      int f  = tid + 256 * j;       // float4 index
      int r  = f >> 3;              // 8 float4 per row
      int c4 = f & 7;
      int gr = m0b + r; if (gr >= M) gr = M - 1;   // clamp (rows not stored)
      const float4 v = *(const float4*)(A + (size_t)gr * K + k0 + c4 * 4);
      As[r][c4 * 4 + 0] = v.x; As[r][c4 * 4 + 1] = v.y;
      As[r][c4 * 4 + 2] = v.z; As[r][c4 * 4 + 3] = v.w;
    }
    // ---- stage B tile 32x64 TRANSPOSED: 512 float4, 2 per thread ----
#pragma unroll
    for (int j = 0; j < 2; ++j) {
      int f  = tid + 256 * j;
      int r  = f >> 4;              // k row (0..31), 16 float4 per row
      int c4 = f & 15;              // n group of 4
      const float4 v = *(const float4*)(Bw + (size_t)(k0 + r) * N + n0 + c4 * 4);
      BsT[c4 * 4 + 0][r] = v.x;
      BsT[c4 * 4 + 1][r] = v.y;
      BsT[c4 * 4 + 2][r] = v.z;
      BsT[c4 * 4 + 3][r] = v.w;
    }
    // ---- prefetch next K tile into cache (global_prefetch_b8) ----
    if (k0 + 32 < K) {
      int r  = tid >> 3;
      int c4 = tid & 7;
      int gr = m0b + r; if (gr >= M) gr = M - 1;
      __builtin_prefetch(A + (size_t)gr * K + k0 + 32 + c4 * 4, 0, 3);
      if (tid < 128) {
        int rb  = tid >> 2;
        int cb4 = tid & 3;
        __builtin_prefetch(Bw + (size_t)(k0 + 32 + rb) * N + n0 + cb4 * 16, 0, 3);
      }
    }
    __syncthreads();

    const int mrow0 = wr * 32 + l16;
    const int mrow1 = wr * 32 + 16 + l16;
    const int ncol0 = wc * 32 + l16;
    const int ncol1 = wc * 32 + 16 + l16;
#pragma unroll
    for (int kk = 0; kk < 32; kk += 4) {
      const int ko = kk + 2 * half;
      v2f a0, a1, b0, b1;
      a0.x = As [mrow0][ko]; a0.y = As [mrow0][ko + 1];
      a1.x = As [mrow1][ko]; a1.y = As [mrow1][ko + 1];
      b0.x = BsT[ncol0][ko]; b0.y = BsT[ncol0][ko + 1];
      b1.x = BsT[ncol1][ko]; b1.y = BsT[ncol1][ko + 1];
      acc[0] = wmma_f32(a0, b0, acc[0]);
      acc[1] = wmma_f32(a0, b1, acc[1]);
      acc[2] = wmma_f32(a1, b0, acc[2]);
      acc[3] = wmma_f32(a1, b1, acc[3]);
    }
    __syncthreads();
  }

#pragma unroll
  for (int mt = 0; mt < 2; ++mt) {
#pragma unroll
    for (int nt = 0; nt < 2; ++nt) {
      const v8f a = acc[mt * 2 + nt];
      const int m0 = m0b + wr * 32 + mt * 16;
      const int c0 = n0 + wc * 32 + nt * 16 + l16;
#pragma unroll
      for (int j = 0; j < 8; ++j) {
        int gr = m0 + j + 8 * half;
        if (gr < M) C[(size_t)gr * N + c0] = a[j];
      }
    }
  }
}

// ---------------------------------------------------------------------------
// Windowed attention: one block per (window, head, batch). 64x64x64 per block.
// qkv layout: [B, T, 1536] (q | k | v). Output layout: [B, T, H*64] fp32.
// V is staged transposed (VsT[d][t]) so PV B-fragments are contiguous pairs.
// ---------------------------------------------------------------------------
__global__ __launch_bounds__(256) void win_attn(
    const float* __restrict__ qkv, float* __restrict__ attn_out)
{
  __shared__ float Qs [64][65];   // reused as P (probabilities) after S phase
  __shared__ float Ks [64][65];   // Ks[t][d]
  __shared__ float VsT[64][65];   // VsT[d][t] = V[t][d]

  const int tid  = threadIdx.x;
  const int lane = tid & 31;
  const int wave = tid >> 5;
  const int half = lane >> 4;
  const int l16  = lane & 15;

  const int w = blockIdx.x;
  const int h = blockIdx.y;
  const int b = blockIdx.z;
  const int t0 = 1 + w * 64;

  // ---- load Q,K,V 64x64 tiles (1024 float4 each, 4 per thread) ----
#pragma unroll
  for (int j = 0; j < 4; ++j) {
    int f  = tid + 256 * j;
    int r  = f >> 4;              // token row, 16 float4 per row
    int c4 = f & 15;
    size_t rowbase = ((size_t)b * T_ + t0 + r) * QKVN + h * HD + c4 * 4;
    float4 q4 = *(const float4*)(qkv + rowbase + 0);
    float4 k4 = *(const float4*)(qkv + rowbase + 512);
    float4 v4 = *(const float4*)(qkv + rowbase + 1024);
    Qs[r][c4*4+0] = q4.x; Qs[r][c4*4+1] = q4.y; Qs[r][c4*4+2] = q4.z; Qs[r][c4*4+3] = q4.w;
    Ks[r][c4*4+0] = k4.x; Ks[r][c4*4+1] = k4.y; Ks[r][c4*4+2] = k4.z; Ks[r][c4*4+3] = k4.w;
    VsT[c4*4+0][r] = v4.x; VsT[c4*4+1][r] = v4.y;
    VsT[c4*4+2][r] = v4.z; VsT[c4*4+3][r] = v4.w;
  }
  __syncthreads();

  // ---- S = Q K^T * scale : 16 tiles of 16x16, 2 per wave (same tm) ----
  v8f accs[2];
#pragma unroll
  for (int rep = 0; rep < 2; ++rep) {
    int ti = wave * 2 + rep;
    int tm = ti >> 2, tn = ti & 3;
    v8f acc = {0.f, 0.f, 0.f, 0.f, 0.f, 0.f, 0.f, 0.f};
#pragma unroll
    for (int kk = 0; kk < 64; kk += 4) {
      const int ko = kk + 2 * half;
      v2f a, bb;
      a.x  = Qs[tm * 16 + l16][ko];
      a.y  = Qs[tm * 16 + l16][ko + 1];
      bb.x = Ks[tn * 16 + l16][ko];   // B[k,n] = K[n,k] (K^T)
      bb.y = Ks[tn * 16 + l16][ko + 1];
      acc = wmma_f32(a, bb, acc);
    }
    accs[rep] = acc;
  }
  __syncthreads();   // all reads of Qs done; safe to reuse as P

#pragma unroll
  for (int rep = 0; rep < 2; ++rep) {
    int ti = wave * 2 + rep;
    int tm = ti >> 2, tn = ti & 3;
#pragma unroll
    for (int j = 0; j < 8; ++j)
      Qs[tm * 16 + j + 8 * half][tn * 16 + l16] = accs[rep][j] * SCALE;
  }
  __syncthreads();

  // ---- row softmax (64 rows, one lane each) ----
  if (tid < 64) {
    float m = -1e30f;
#pragma unroll 8
    for (int c = 0; c < 64; ++c) m = fmaxf(m, Qs[tid][c]);
    float s = 0.f;
#pragma unroll 8
    for (int c = 0; c < 64; ++c) { float e = __expf(Qs[tid][c] - m); Qs[tid][c] = e; s += e; }
    float inv = 1.0f / s;
#pragma unroll 8
    for (int c = 0; c < 64; ++c) Qs[tid][c] *= inv;
  }
  __syncthreads();

  // ---- O = P V, store to (b, t, h, d) flattened as [B*T, 512] ----
#pragma unroll
  for (int rep = 0; rep < 2; ++rep) {
    int ti = wave * 2 + rep;
    int tm = ti >> 2, tn = ti & 3;
    v8f acc = {0.f, 0.f, 0.f, 0.f, 0.f, 0.f, 0.f, 0.f};
#pragma unroll
    for (int kk = 0; kk < 64; kk += 4) {
      const int ko = kk + 2 * half;
      v2f a, bb;
      a.x  = Qs[tm * 16 + l16][ko];         // P
      a.y  = Qs[tm * 16 + l16][ko + 1];
      bb.x = VsT[tn * 16 + l16][ko];        // B[k,n] = V[k,n] = VsT[n][k]
      bb.y = VsT[tn * 16 + l16][ko + 1];
      acc = wmma_f32(a, bb, acc);
    }
#pragma unroll
    for (int j = 0; j < 8; ++j) {
      size_t row = (size_t)b * T_ + t0 + tm * 16 + j + 8 * half;
      attn_out[row * Dm + h * HD + tn * 16 + l16] = acc[j];
    }
  }
}

// ---------------------------------------------------------------------------
// Global attention for token 0: one block per (head, batch). VALU-only.
// ---------------------------------------------------------------------------
__global__ __launch_bounds__(256) void attn0_kernel(
    const float* __restrict__ qkv, float* __restrict__ scores,
    float* __restrict__ attn_out)
{
  __shared__ float q0[64];
  __shared__ float red[256];

  const int tid = threadIdx.x;
  const int h = blockIdx.x, b = blockIdx.y;
  float* sc = scores + ((size_t)b * H_ + h) * T_;
  const size_t base = (size_t)b * T_ * QKVN;

  if (tid < 64) q0[tid] = qkv[base + h * HD + tid];
  __syncthreads();

  // pass 1: scores + max
  float lmax = -1e30f;
  for (int t = tid; t < T_; t += 256) {
    const float* kp = qkv + base + (size_t)t * QKVN + 512 + h * HD;
    float s = 0.f;
#pragma unroll 8
    for (int i = 0; i < 64; ++i) s += q0[i] * kp[i];
    s *= SCALE;
    sc[t] = s;
    lmax = fmaxf(lmax, s);
  }
  red[tid] = lmax; __syncthreads();
  for (int off = 128; off > 0; off >>= 1) {
    if (tid < off) red[tid] = fmaxf(red[tid], red[tid + off]);
    __syncthreads();
  }
  const float gmax = red[0]; __syncthreads();

  // pass 2: exp + sum
  float lsum = 0.f;
  for (int t = tid; t < T_; t += 256) {
    float e = __expf(sc[t] - gmax);
    sc[t] = e;
    lsum += e;
  }
  red[tid] = lsum; __syncthreads();
  for (int off = 128; off > 0; off >>= 1) {
    if (tid < off) red[tid] += red[tid + off];
    __syncthreads();
  }
  const float inv = 1.0f / red[0]; __syncthreads();

  // pass 3: out[d] = sum_t p_t * v[t][d]; 4 partial stripes per dim
  const int dim = tid & 63;
  const int qrt = tid >> 6;
  float pacc = 0.f;
  for (int t = qrt; t < T_; t += 4)
    pacc += sc[t] * qkv[base + (size_t)t * QKVN + 1024 + h * HD + dim];
  red[tid] = pacc; __syncthreads();
  if (tid < 64) {
    float o = (red[tid] + red[tid + 64] + red[tid + 128] + red[tid + 192]) * inv;
    attn_out[(size_t)b * T_ * Dm + h * HD + tid] = o;
  }
}

// ---------------------------------------------------------------------------
extern "C" void kernel_launch(void* const* d_in, const int* in_sizes, int n_in,
                              void* d_out, int out_size, void* d_ws, size_t ws_size,
                              hipStream_t stream)
{
  (void)in_sizes; (void)n_in; (void)out_size; (void)ws_size;

  const float* x     = (const float*)d_in[0];   // [B,T,512]
  const float* w_qkv = (const float*)d_in[1];   // [512,1536]
  const float* w_out = (const float*)d_in[2];   // [512,512]
  float* out = (float*)d_out;                   // [B,T,512]

  // Workspace carve-up (~270 MB total):
  char* ws = (char*)d_ws;
  const size_t qkv_bytes  = (size_t)B_ * T_ * QKVN * sizeof(float);  // 201 MB
  const size_t attn_bytes = (size_t)B_ * T_ * Dm * sizeof(float);    //  67 MB
  float* qkv    = (float*)ws;
  float* attn   = (float*)(ws + qkv_bytes);
  float* scores = (float*)(ws + qkv_bytes + attn_bytes);             //   1 MB

  const int M = B_ * T_;   // 32776
  dim3 blk(256);

  // 1) qkv = x @ w_qkv            [32776,512] x [512,1536]
  dim3 g1((M + 127) / 128, QKVN / 64);
  gemm_f32_wmma<<<g1, blk, 0, stream>>>(x, w_qkv, qkv, M, QKVN, Dm);

  // 2) windowed attention on tokens 1..4096
  dim3 g2(NW, H_, B_);
  win_attn<<<g2, blk, 0, stream>>>(qkv, attn);

  // 3) global attention for token 0
  dim3 g3(H_, B_);
  attn0_kernel<<<g3, blk, 0, stream>>>(qkv, scores, attn);

  // 4) out = attn @ w_out         [32776,512] x [512,512]
  dim3 g4((M + 127) / 128, Dm / 64);
  gemm_f32_wmma<<<g4, blk, 0, stream>>>(attn, w_out, out, M, Dm, Dm);
}